// TimeAwareMultiHeadAttention_7430293422820
// MI455X (gfx1250) — compile-verified
//
#include <hip/hip_runtime.h>
#include <hip/hip_bf16.h>

// ---------------------------------------------------------------------------
// TimeAwareMultiHeadAttention (TiSASRec-style), B=16 L=256 H=128 NH=2 D=64
//
// Refactoring: never materialize tK/tV (would be 2x537MB + 69 GFLOP).
//   logits_time[b,h,l,m] = tm[b,l,m,:] . qA[b,l,h,:],
//       qA[b,l,h,c] = sum_d Q[b,l,h,d] * WA1[h*64+d, c]
//   out_time[b,l,h,:]  = s[b,l,h,:] @ WA2[h*64:(h+1)*64, :]^T,
//       s[b,l,h,c] = sum_m probs[b,h,l,m] * tm[b,l,m,c]
// => single pass over time_matrix (537 MB), staged per-(b,l) in 128 KB LDS
//    via the Tensor Data Mover (TENSOR_LOAD_TO_LDS, TENSORcnt).
// HBM-bound: ~537MB / 23.3 TB/s ~= 23 us floor; GEMMs use
// V_WMMA_F32_16X16X4_F32 (fp32 kept end-to-end like the reference).
// ---------------------------------------------------------------------------

#define BATCH 16
#define SEQL  256
#define HID   128
#define NHEAD 2
#define HDIM  64
#define ROWS  (BATCH * SEQL)          // 4096
#define NEGF  (-4294967295.0f)        // -2^32 + 1 (rounds to -2^32 in f32)

// workspace layout (floats)
#define Q_OFF  0
#define K_OFF  (ROWS * HID)           // 524288
#define V_OFF  (2 * ROWS * HID)       // 1048576
#define QA_OFF (3 * ROWS * HID)       // 1572864 ; qA is (4096, 256)

typedef __attribute__((ext_vector_type(2))) float v2f;
typedef __attribute__((ext_vector_type(8))) float v8f;
typedef __attribute__((ext_vector_type(4))) unsigned int u32x4;
typedef __attribute__((ext_vector_type(8))) int i32x8;
typedef __attribute__((ext_vector_type(4))) int i32x4;

// ---------------------------------------------------------------------------
// Kernel 1: Q/K/V = x @ W^T + b  via fp32 WMMA 16x16x4.
// One wave (32 threads) computes one 16x16 output tile.
// A 16x4 frag: lanes 0-15 hold M=lane, VGPR0=K0,VGPR1=K1; lanes 16-31 K2/K3.
// B 4x16 frag: lanes 0-15 hold N=lane, VGPR0=K0,VGPR1=K1; lanes 16-31 K2/K3.
// For B = W^T, Bmat[k][n] = W[n][k] -> contiguous float2 per lane.
// ---------------------------------------------------------------------------
__global__ __launch_bounds__(32)
void proj_qkv_kernel(const float* __restrict__ x,
                     const float* __restrict__ Qw, const float* __restrict__ Qb,
                     const float* __restrict__ Kw, const float* __restrict__ Kb,
                     const float* __restrict__ Vw, const float* __restrict__ Vb,
                     float* __restrict__ ws) {
  const int tileM = blockIdx.x;            // 0..255
  const int tileN = blockIdx.y;            // 0..7
  const int which = blockIdx.z;            // 0=Q 1=K 2=V
  const float* W    = (which == 0) ? Qw : (which == 1) ? Kw : Vw;
  const float* bias = (which == 0) ? Qb : (which == 1) ? Kb : Vb;
  float* out = ws + (size_t)which * (ROWS * HID);

  const int lane = threadIdx.x;            // 0..31 (wave32)
  const int half = lane >> 4;              // 0/1 -> K pair select / M-half
  const int l15  = lane & 15;
  const int m0 = tileM * 16, n0 = tileN * 16;

  v8f acc = {};
  for (int k0 = 0; k0 < HID; k0 += 4) {
    const float* ap = x + (size_t)(m0 + l15) * HID + (k0 + 2 * half);
    v2f a; a.x = ap[0]; a.y = ap[1];
    const float* bp = W + (size_t)(n0 + l15) * HID + (k0 + 2 * half); // W^T access
    v2f bfrag; bfrag.x = bp[0]; bfrag.y = bp[1];
    acc = __builtin_amdgcn_wmma_f32_16x16x4_f32(false, a, false, bfrag,
                                                (short)0, acc, false, false);
  }
  const float bv = bias[n0 + l15];
  #pragma unroll
  for (int r = 0; r < 8; ++r) {
    const int row = m0 + r + half * 8;     // C/D: VGPR r -> M=r (+8 for hi half)
    out[(size_t)row * HID + n0 + l15] = acc[r] + bv;
  }
}

// ---------------------------------------------------------------------------
// Kernel 2: qA[row, h*128+c] = sum_d Q[row, h*64+d] * WA1[h*64+d, c]
// B is NOT transposed here (K index = row of WA1).
// ---------------------------------------------------------------------------
__global__ __launch_bounds__(32)
void proj_qa_kernel(const float* __restrict__ ws,    // Q at offset 0
                    const float* __restrict__ WA1,
                    float* __restrict__ qa) {
  const int tileM = blockIdx.x;            // 0..255
  const int tileN = blockIdx.y;            // 0..7
  const int head  = blockIdx.z;            // 0..1
  const float* Q = ws + Q_OFF;

  const int lane = threadIdx.x;
  const int half = lane >> 4;
  const int l15  = lane & 15;
  const int m0 = tileM * 16, n0 = tileN * 16;

  v8f acc = {};
  for (int k0 = 0; k0 < HDIM; k0 += 4) {
    const float* ap = Q + (size_t)(m0 + l15) * HID + head * HDIM + (k0 + 2 * half);
    v2f a; a.x = ap[0]; a.y = ap[1];
    const int krow = head * HDIM + k0 + 2 * half;
    v2f bfrag;
    bfrag.x = WA1[(size_t)krow * HID + n0 + l15];
    bfrag.y = WA1[(size_t)(krow + 1) * HID + n0 + l15];
    acc = __builtin_amdgcn_wmma_f32_16x16x4_f32(false, a, false, bfrag,
                                                (short)0, acc, false, false);
  }
  #pragma unroll
  for (int r = 0; r < 8; ++r) {
    const int row = m0 + r + half * 8;
    qa[(size_t)row * (NHEAD * HID) + head * HID + n0 + l15] = acc[r];
  }
}

// ---------------------------------------------------------------------------
// Kernel 3: one workgroup (256 threads = 8 waves) per (b,l).
// Stage tm[b,l,:,:] (256x128 f32 = 128 KB) in LDS via the Tensor Data Mover;
// compute logits, softmax, s = probs @ tm, out = probs @ V + s @ WA2_h^T.
// 136.7 KB LDS -> 2 WGs/WGP; the co-resident WG's TDM DMA overlaps compute.
// ---------------------------------------------------------------------------
__global__ __launch_bounds__(256)
void attn_kernel(const float* __restrict__ time_matrix,
                 const unsigned char* __restrict__ time_mask,  // (B,L) bool
                 const unsigned char* __restrict__ attn_mask,  // (L,L) bool
                 const float* __restrict__ ws,
                 const float* __restrict__ WA2,
                 float* __restrict__ out) {
  __shared__ __align__(16) float sh_tm[SEQL * HID];  // 32768 floats = 128 KB
  __shared__ float sh_probs[NHEAD * SEQL];           // 512
  __shared__ float sh_q[HID];                        // 128
  __shared__ float sh_qa[NHEAD * HID];               // 256
  __shared__ float sh_s[NHEAD * HID];                // 256
  __shared__ float sh_red[256];                      // reduction scratch

  const int bl = blockIdx.x;                // 0..4095
  const int b  = bl >> 8;
  const int l  = bl & (SEQL - 1);
  const int t  = threadIdx.x;               // 0..255

  const float* Q  = ws + Q_OFF;
  const float* K  = ws + K_OFF;
  const float* V  = ws + V_OFF;
  const float* qa = ws + QA_OFF;

  // ---- stage tm[b,l,:,:] into LDS ------------------------------------------
  const float* tmg = time_matrix + (size_t)bl * (SEQL * HID);
#if __has_builtin(__builtin_amdgcn_tensor_load_to_lds) && \
    __has_builtin(__builtin_amdgcn_s_wait_tensorcnt)
  // Tensor Data Mover: one DMA descriptor (2D tile 256 rows x 128 ch, f32),
  // issued by wave 0 only (TDM ignores EXEC; one issue per executing wave).
  if (t < 32) {
    const unsigned long long gaddr = (unsigned long long)(const void*)tmg;
    const unsigned int lds_off = (unsigned int)(unsigned long long)(const void*)sh_tm;
    u32x4 g0;
    g0[0] = 1u;                                     // count=1, user descriptor
    g0[1] = lds_off;                                // D#.lds_addr  [63:32]
    g0[2] = (unsigned int)(gaddr & 0xFFFFFFFFull);  // D#.global_addr[95:64]
    g0[3] = (unsigned int)((gaddr >> 32) & 0x1FFFFFFull)  // addr bits 56:32
            | (2u << 30);                           // D#.type = 2 ("image")
    i32x8 g1;
    g1[0] = (int)(2u << 16);          // data_size=2 -> 4 bytes; no multicast
    g1[1] = (int)(128u << 16);        // tensor_dim0[15:0]=128 @ bits 63:48
    g1[2] = (int)(256u << 16);        // tensor_dim1[15:0]=256 @ bits 95:80
    g1[3] = (int)(128u << 16);        // tile_dim0=128         @ bits 127:112
    g1[4] = (int)256u;                // tile_dim1=256         @ bits 143:128
    g1[5] = (int)128u;                // tensor_dim0_stride=128 (low 32 bits)
    g1[6] = 0;
    g1[7] = 0;
    i32x4 gz4 = {};                   // groups 2/3 unused (2D tensor)
    i32x8 gz8 = {};                   // 6-arg toolchain form: extra group
    __builtin_amdgcn_tensor_load_to_lds(g0, g1, gz4, gz4, gz8, 0);
    __builtin_amdgcn_s_wait_tensorcnt(0);           // wave 0 gates the barrier
  }
#else
  for (int i = t * 4; i < SEQL * HID; i += 256 * 4) {
    *(float4*)(sh_tm + i) = *(const float4*)(tmg + i);
  }
#endif
  if (t < HID) sh_q[t] = Q[(size_t)bl * HID + t];
  sh_qa[t] = qa[(size_t)bl * (NHEAD * HID) + t];
  __syncthreads();

  // ---- logits[h][m] = (Q.K + tm.qA) / sqrt(D), masked ----------------------
  const bool qmasked = (time_mask[bl] != 0);
  const int h  = t >> 7;                    // 0..1
  const int mi = t & 127;
  #pragma unroll
  for (int rep = 0; rep < 2; ++rep) {
    const int m = mi + rep * 128;
    float acc = 0.0f;
    const float* kp = K + ((size_t)b * SEQL + m) * HID + h * HDIM;
    const float* qp = sh_q + h * HDIM;
    for (int d = 0; d < HDIM; ++d) acc += qp[d] * kp[d];
    const float* tp = sh_tm + m * HID;
    const float* ap = sh_qa + h * HID;
    for (int c = 0; c < HID; ++c) acc += ap[c] * tp[c];
    acc *= 0.125f;                          // 1/sqrt(64)
    const bool masked = qmasked || (attn_mask[l * SEQL + m] != 0);
    sh_probs[h * SEQL + m] = masked ? NEGF : acc;
  }
  __syncthreads();

  // ---- softmax over m per head (128 threads per head reduce 256 vals) ------
  const float v0 = sh_probs[h * SEQL + mi];
  const float v1 = sh_probs[h * SEQL + mi + 128];
  sh_red[t] = fmaxf(v0, v1);
  __syncthreads();
  for (int off = 64; off >= 1; off >>= 1) {
    if (mi < off) sh_red[h * 128 + mi] = fmaxf(sh_red[h * 128 + mi],
                                               sh_red[h * 128 + mi + off]);
    __syncthreads();
  }
  const float rowmax = sh_red[h * 128];
  __syncthreads();
  const float e0 = expf(v0 - rowmax);
  const float e1 = expf(v1 - rowmax);
  sh_red[t] = e0 + e1;
  __syncthreads();
  for (int off = 64; off >= 1; off >>= 1) {
    if (mi < off) sh_red[h * 128 + mi] += sh_red[h * 128 + mi + off];
    __syncthreads();
  }
  const float inv = 1.0f / sh_red[h * 128];
  sh_probs[h * SEQL + mi]       = e0 * inv;
  sh_probs[h * SEQL + mi + 128] = e1 * inv;
  __syncthreads();

  // ---- s[h][c] = sum_m probs[h][m] * tm[m][c] ------------------------------
  {
    const int c = mi;                       // thread t -> (h, c)
    float acc = 0.0f;
    for (int m = 0; m < SEQL; ++m) acc += sh_probs[h * SEQL + m] * sh_tm[m * HID + c];
    sh_s[t] = acc;
  }
  __syncthreads();

  // ---- out[b,l,h*64+d] = sum_m probs*V + sum_c s*WA2[h*64+d, c] ------------
  if (t < NHEAD * HDIM) {
    const int hh = t >> 6;                  // 0..1
    const int d  = t & 63;
    float accp = 0.0f;
    const float* vp = V + (size_t)b * SEQL * HID + hh * HDIM + d;
    for (int m = 0; m < SEQL; ++m) accp += sh_probs[hh * SEQL + m] * vp[(size_t)m * HID];
    float acct = 0.0f;
    const float* wp = WA2 + (size_t)(hh * HDIM + d) * HID;
    const float* sp = sh_s + hh * HID;
    for (int c = 0; c < HID; ++c) acct += sp[c] * wp[c];
    out[(size_t)bl * HID + hh * HDIM + d] = accp + acct;
  }
}

// ---------------------------------------------------------------------------
extern "C" void kernel_launch(void* const* d_in, const int* in_sizes, int n_in,
                              void* d_out, int out_size, void* d_ws, size_t ws_size,
                              hipStream_t stream) {
  (void)in_sizes; (void)n_in; (void)out_size; (void)ws_size;
  const float*         item  = (const float*)d_in[0];
  const unsigned char* tmask = (const unsigned char*)d_in[1];   // bool (1B)
  const unsigned char* amask = (const unsigned char*)d_in[2];   // bool (1B)
  const float*         tmat  = (const float*)d_in[3];
  const float* Qw  = (const float*)d_in[4];
  const float* Qb  = (const float*)d_in[5];
  const float* Kw  = (const float*)d_in[6];
  const float* Kb  = (const float*)d_in[7];
  const float* Vw  = (const float*)d_in[8];
  const float* Vb  = (const float*)d_in[9];
  const float* WA1 = (const float*)d_in[10];
  const float* WA2 = (const float*)d_in[11];
  float* ws  = (float*)d_ws;     // needs (3*ROWS*HID + ROWS*256)*4 = 10 MB
  float* out = (float*)d_out;

  dim3 gqkv(ROWS / 16, HID / 16, 3);
  proj_qkv_kernel<<<gqkv, 32, 0, stream>>>(item, Qw, Qb, Kw, Kb, Vw, Vb, ws);

  dim3 gqa(ROWS / 16, HID / 16, NHEAD);
  proj_qa_kernel<<<gqa, 32, 0, stream>>>(ws, WA1, ws + QA_OFF);

  attn_kernel<<<ROWS, 256, 0, stream>>>(tmat, tmask, amask, ws, WA2, out);
}